// FisherLoss_89326729822506
// MI455X (gfx1250) — compile-verified
//
#include <hip/hip_runtime.h>

// FisherLoss (Sw branch): dist_i = ||x_i||^2 + ||c_lab||^2 - 2 x_i.c_lab ; clip ; mean.
// One pass over memory (~2 MB total) -> latency-bound on MI455X (23.3 TB/s => ~90ns
// of pure bandwidth). Strategy: maximize memory-level parallelism. One workgroup
// (8 waves) per 16-row tile, waves split the 512-dim K range 8 ways (64 dims each),
// fully unrolled => deep load clauses. Dot products run on the CDNA5 matrix pipe via
// V_WMMA_F32_16X16X4_F32 (f32 A/B -> full precision). Partials combined through LDS
// in a fixed order (bit-deterministic, no float atomics).

#define FEAT 512
#define BATCH 512
#define TILE 16
#define NTILES (BATCH / TILE)   // 32 tiles, one workgroup each
#define KSPLIT 8                // waves per workgroup, each owns FEAT/KSPLIT dims
#define KCHUNK (FEAT / KSPLIT)  // 64

typedef __attribute__((ext_vector_type(2))) float v2f;
typedef __attribute__((ext_vector_type(8))) float v8f;

__device__ __forceinline__ float pick8(v8f v, int j) {
    // select diagonal element j (0..7) from the 8-VGPR accumulator
    float r = v[0];
    r = (j == 1) ? v[1] : r;
    r = (j == 2) ? v[2] : r;
    r = (j == 3) ? v[3] : r;
    r = (j == 4) ? v[4] : r;
    r = (j == 5) ? v[5] : r;
    r = (j == 6) ? v[6] : r;
    r = (j == 7) ? v[7] : r;
    return r;
}

__global__ void __launch_bounds__(256)
fisher_sw_wmma_kernel(const float* __restrict__ x,
                      const int* __restrict__ labels,
                      const float* __restrict__ centers,
                      float* __restrict__ partial) {
    __shared__ float s[KSPLIT][TILE][3];   // per-wave partial (xx, cc, xc) per row

    const int tile = (int)blockIdx.x;          // 0..31
    const int wv   = (int)(threadIdx.x >> 5);  // 0..7 : K-chunk owned by this wave
    const int lane = (int)(threadIdx.x & 31);
    const int rowInTile = lane & 15;
    const int row  = tile * TILE + rowInTile;  // 0..511
    const int koff = (lane >> 4) << 1;         // 0 or 2 (A/B K sub-offset)
    const int kbase = wv * KCHUNK;

    // Lane l holds, for both the A tile (16x4, rows in lanes) and the B tile
    // (4x16, columns in lanes), elements [row, k+koff] and [row, k+koff+1]
    // of the source matrix -> one float2 load feeds either operand role.
    const float* __restrict__ px = x + (size_t)row * FEAT + kbase + koff;
    const int lab = labels[row];
    const float* __restrict__ pc = centers + (size_t)lab * FEAT + kbase + koff;

    v8f axc = {};  // diag -> sum x.c   over this wave's K-chunk
    v8f axx = {};  // diag -> sum x.x
    v8f acc = {};  // diag -> sum c.c

#pragma unroll
    for (int k = 0; k < KCHUNK; k += 4) {
        v2f a = *(const v2f*)(px + k);
        v2f b = *(const v2f*)(pc + k);
        // 8 args: (neg_a, A, neg_b, B, c_mod, C, reuse_a, reuse_b)
        axc = __builtin_amdgcn_wmma_f32_16x16x4_f32(false, a, false, b,
                                                    (short)0, axc, false, false);
        axx = __builtin_amdgcn_wmma_f32_16x16x4_f32(false, a, false, a,
                                                    (short)0, axx, false, false);
        acc = __builtin_amdgcn_wmma_f32_16x16x4_f32(false, b, false, b,
                                                    (short)0, acc, false, false);
    }

    // Diagonal element r of 16x16 f32 C/D: VGPR r%8, lane r (r<8) or lane r+16 (r>=8).
    int j = -1;
    if (lane < 8)        j = lane;        // rows 0..7
    else if (lane >= 24) j = lane - 24;   // rows 8..15
    if (j >= 0) {
        const int r = (lane < 8) ? lane : (lane - 16);  // row index 0..15
        s[wv][r][0] = pick8(axx, j);
        s[wv][r][1] = pick8(acc, j);
        s[wv][r][2] = pick8(axc, j);
    }
    __syncthreads();

    // Wave 0: fixed-order combine across the 8 K-chunks, per-row clip, reduce.
    if (threadIdx.x < 32) {
        float val = 0.0f;
        if (lane < 16) {
            float xx = 0.0f, cc = 0.0f, xc = 0.0f;
#pragma unroll
            for (int w = 0; w < KSPLIT; ++w) {
                xx += s[w][lane][0];
                cc += s[w][lane][1];
                xc += s[w][lane][2];
            }
            float dist = xx + cc - 2.0f * xc;
            dist = fminf(fmaxf(dist, 1e-12f), 1e12f);  // per-row clip before mean
            val = dist;
        }
#pragma unroll
        for (int off = 16; off > 0; off >>= 1)
            val += __shfl_xor(val, off, 32);
        if (lane == 0)
            partial[tile] = val;
    }
}

__global__ void fisher_sw_finalize_kernel(const float* __restrict__ partial,
                                          float* __restrict__ out) {
    const int lane = (int)(threadIdx.x & 31);
    float v = partial[lane];  // exactly NTILES==32 partials
#pragma unroll
    for (int off = 16; off > 0; off >>= 1)
        v += __shfl_xor(v, off, 32);
    if (lane == 0)
        out[0] = v * (1.0f / (float)BATCH);
}

extern "C" void kernel_launch(void* const* d_in, const int* in_sizes, int n_in,
                              void* d_out, int out_size, void* d_ws, size_t ws_size,
                              hipStream_t stream) {
    (void)in_sizes; (void)n_in; (void)out_size; (void)ws_size;
    const float* x        = (const float*)d_in[0];
    const int*   labels   = (const int*)d_in[1];
    const float* centers  = (const float*)d_in[2];
    // d_in[3] (feature_mean) is unused by the returned loss.
    float* partial = (float*)d_ws;   // 32 floats of scratch
    float* out     = (float*)d_out;

    // 32 workgroups x 256 threads = 256 waves: 8x the memory-level parallelism
    // of a one-wave-per-tile mapping; each wave covers 64 of the 512 dims.
    fisher_sw_wmma_kernel<<<dim3(NTILES), dim3(256), 0, stream>>>(
        x, labels, centers, partial);
    fisher_sw_finalize_kernel<<<dim3(1), dim3(32), 0, stream>>>(partial, out);
}